// AttentionNeNet_30554397343923
// MI455X (gfx1250) — compile-verified
//
#include <hip/hip_runtime.h>
#include <hip/hip_bf16.h>

// ---------------------------------------------------------------------------
// Problem constants (from reference)
// ---------------------------------------------------------------------------
#define IN_SIZE   256
#define OUT_SIZE  64
#define N_NODES   512
#define FAN_IN    32
#define T_LEN     2048
#define D_LEN     832                 // IN_SIZE + N_NODES + OUT_SIZE
#define NCOLS     1024                // 512 K columns + 512 V columns

typedef __attribute__((ext_vector_type(2))) float v2f;
typedef __attribute__((ext_vector_type(8))) float v8f;

// ---------------------------------------------------------------------------
// Phase 0a: build transposed activation matrix
//   AsT[d][t] = actives[t+1][d] for t < 2047, 0 for t == 2047
// (transposed so GEMM A-fragment loads are lane-contiguous: lanes 0..15 read
//  16 consecutive floats of one k-row)
// ---------------------------------------------------------------------------
__global__ void prep_ast_kernel(const float* __restrict__ actives,
                                float* __restrict__ AsT) {
    int i = blockIdx.x * blockDim.x + threadIdx.x;     // i in [0, 2048*832)
    int t = i / D_LEN;
    int d = i - t * D_LEN;
    float v = (t < T_LEN - 1) ? actives[i + D_LEN] : 0.0f;  // coalesced read
    AsT[(size_t)d * T_LEN + t] = v;                          // scattered write (one-time)
}

// ---------------------------------------------------------------------------
// Phase 0b: zero the dense scattered-weight matrix B[832][1024]
// ---------------------------------------------------------------------------
__global__ void zero_b_kernel(float* __restrict__ B) {
    int i = blockIdx.x * blockDim.x + threadIdx.x;     // i in [0, 832*1024)
    B[i] = 0.0f;
}

// ---------------------------------------------------------------------------
// Phase 0c: scatter Wk/Wv into dense B. One thread per node => deterministic,
// handles duplicate gather indices within a node via sequential +=.
//   B[idx][n]       += W[n][j][1]   (k weights)
//   B[idx][512 + n] += W[n][j][2]   (v weights)
// ---------------------------------------------------------------------------
__global__ void scatter_w_kernel(const float* __restrict__ weights,
                                 const int* __restrict__ in_idxs,
                                 float* __restrict__ B) {
    int n = blockIdx.x * blockDim.x + threadIdx.x;
    if (n >= N_NODES) return;
    for (int j = 0; j < FAN_IN; ++j) {
        int idx = in_idxs[n * FAN_IN + j];
        float wk = weights[(n * FAN_IN + j) * 3 + 1];
        float wv = weights[(n * FAN_IN + j) * 3 + 2];
        B[idx * NCOLS + n]           += wk;
        B[idx * NCOLS + N_NODES + n] += wv;
    }
}

// ---------------------------------------------------------------------------
// Phase 1: dense fp32 WMMA GEMM  KV[2048][1024] = A[2048][832] @ B[832][1024]
// (A supplied transposed as AsT[832][2048]).
//
// Each wave computes a 32x32 output tile as a 2x2 grid of 16x16 WMMA tiles
// (V_WMMA_F32_16X16X4_F32): per k-step, 2 A-fragments + 2 B-fragments are
// loaded (8 x b32, each lane-contiguous) and reused across 4 WMMAs
// => 2 VMEM issues per WMMA instead of 3, all 64B-segment coalesced.
//
// VGPR layouts (ISA 7.12.2, 32-bit data):
//   A 16x4:  lane L holds row M=L%16; VGPR0 = K = 2*(L/16), VGPR1 = K+1
//   B 4x16:  lane L holds col N=L%16; VGPR0 = K = 2*(L/16), VGPR1 = K+1
//   C/D 16x16: VGPR r: lanes 0-15 -> M=r, lanes 16-31 -> M=r+8; N = L%16
// ---------------------------------------------------------------------------
__global__ void __launch_bounds__(128)
gemm_kv_kernel(const float* __restrict__ AsT,
               const float* __restrict__ B,
               float* __restrict__ KV) {
    const int wave = threadIdx.x >> 5;   // 0..3 -> 2x2 wave grid in the block
    const int lane = threadIdx.x & 31;
    const int col  = lane & 15;          // M (A) / N (B) index within a tile
    const int half = lane >> 4;          // selects K pair

    const int tilesN = NCOLS / 64;       // 16: block covers 64 cols x 64 rows
    const int mBase  = (blockIdx.x / tilesN) * 64 + (wave >> 1) * 32;
    const int nBase  = (blockIdx.x % tilesN) * 64 + (wave & 1) * 32;

    const float* aPtr = AsT + mBase + col;   // + kb*T_LEN (+16 for 2nd M tile)
    const float* bPtr = B   + nBase + col;   // + kb*NCOLS (+16 for 2nd N tile)

    // Prefetch start of operand streams (emits global_prefetch_b8).
    __builtin_prefetch(aPtr, 0, 1);
    __builtin_prefetch(bPtr, 0, 1);

    v8f acc00 = {}, acc01 = {}, acc10 = {}, acc11 = {};
    for (int k = 0; k < D_LEN; k += 4) {
        const int kb = k + 2 * half;
        v2f a0, a1, b0, b1;
        a0.x = aPtr[(size_t)kb * T_LEN];
        a0.y = aPtr[(size_t)(kb + 1) * T_LEN];
        a1.x = aPtr[(size_t)kb * T_LEN + 16];
        a1.y = aPtr[(size_t)(kb + 1) * T_LEN + 16];
        b0.x = bPtr[kb * NCOLS];
        b0.y = bPtr[(kb + 1) * NCOLS];
        b1.x = bPtr[kb * NCOLS + 16];
        b1.y = bPtr[(kb + 1) * NCOLS + 16];
        acc00 = __builtin_amdgcn_wmma_f32_16x16x4_f32(false, a0, false, b0,
                                                      (short)0, acc00, false, false);
        acc01 = __builtin_amdgcn_wmma_f32_16x16x4_f32(false, a0, false, b1,
                                                      (short)0, acc01, false, false);
        acc10 = __builtin_amdgcn_wmma_f32_16x16x4_f32(false, a1, false, b0,
                                                      (short)0, acc10, false, false);
        acc11 = __builtin_amdgcn_wmma_f32_16x16x4_f32(false, a1, false, b1,
                                                      (short)0, acc11, false, false);
    }

    float* d00 = KV + (size_t)mBase * NCOLS + nBase + col;
    #pragma unroll
    for (int r = 0; r < 8; ++r) {
        const int m = r + 8 * half;
        d00[(size_t)m * NCOLS]             = acc00[r];
        d00[(size_t)m * NCOLS + 16]        = acc01[r];
        d00[(size_t)(m + 16) * NCOLS]      = acc10[r];
        d00[(size_t)(m + 16) * NCOLS + 16] = acc11[r];
    }
}

// ---------------------------------------------------------------------------
// Phase 2: sequential node chain. One workgroup, 1024 threads (32 waves).
// lastrow lives in LDS; per node: wave0 computes q/k/v_last from the gathered
// last row, then all threads reduce softmax over 2048 elements of q*K column
// and the weighted V sum, tanh, write back.
// ---------------------------------------------------------------------------
__device__ __forceinline__ float waveMax(float v) {
    #pragma unroll
    for (int m = 16; m > 0; m >>= 1) v = fmaxf(v, __shfl_xor(v, m, 32));
    return v;
}
__device__ __forceinline__ float waveSum(float v) {
    #pragma unroll
    for (int m = 16; m > 0; m >>= 1) v += __shfl_xor(v, m, 32);
    return v;
}

__global__ void __launch_bounds__(1024)
seq_attn_kernel(const float* __restrict__ x,
                const float* __restrict__ weights,
                const int* __restrict__ in_idxs,
                const float* __restrict__ KV,
                float* __restrict__ out) {
    __shared__ float lastrow[D_LEN];
    __shared__ float redA[32];
    __shared__ float redB[32];
    __shared__ float redC[32];
    __shared__ float scal[3];            // q_last, k_last, v_last

    const int tid  = threadIdx.x;
    const int lane = tid & 31;
    const int wid  = tid >> 5;

    // init last row: [x ; zeros]
    if (tid < D_LEN) lastrow[tid] = (tid < IN_SIZE) ? x[tid] : 0.0f;
    __syncthreads();

    for (int ix = 0; ix < N_NODES; ++ix) {
        // --- q_last / k_last / v_last from LDS last row (wave 0) ---
        if (wid == 0) {
            float a  = lastrow[in_idxs[ix * FAN_IN + lane]];
            const float* w = weights + (ix * FAN_IN + lane) * 3;
            float q = a * w[0];
            float k = a * w[1];
            float v = a * w[2];
            q = waveSum(q); k = waveSum(k); v = waveSum(v);
            if (lane == 0) { scal[0] = q; scal[1] = k; scal[2] = v; }
        }
        __syncthreads();
        const float ql = scal[0], kl = scal[1], vl = scal[2];

        // --- two logit/value elements per thread ---
        const int t0 = tid;               // 0..1023  (always a fixed row)
        const int t1 = tid + 1024;        // 1024..2047 (t1==2047 -> last row)
        float l0 = ql * KV[t0 * NCOLS + ix];
        float w0 = KV[t0 * NCOLS + N_NODES + ix];
        float l1, w1;
        if (t1 == T_LEN - 1) { l1 = ql * kl; w1 = vl; }
        else                 { l1 = ql * KV[t1 * NCOLS + ix];
                               w1 = KV[t1 * NCOLS + N_NODES + ix]; }

        // --- block max ---
        float m = waveMax(fmaxf(l0, l1));
        if (lane == 0) redA[wid] = m;
        __syncthreads();
        if (wid == 0) {
            float t = waveMax(redA[lane]);
            if (lane == 0) redA[0] = t;
        }
        __syncthreads();
        const float M = redA[0];

        // --- block sum of exp and exp*v ---
        float e0 = __expf(l0 - M);
        float e1 = __expf(l1 - M);
        float se = waveSum(e0 + e1);
        float sv = waveSum(e0 * w0 + e1 * w1);
        if (lane == 0) { redB[wid] = se; redC[wid] = sv; }
        __syncthreads();
        if (wid == 0) {
            float a = waveSum(redB[lane]);
            float b = waveSum(redC[lane]);
            if (lane == 0) {
                float o = tanhf(b / a);
                lastrow[IN_SIZE + ix] = o;
                if (ix >= N_NODES - OUT_SIZE) out[ix - (N_NODES - OUT_SIZE)] = o;
            }
        }
        __syncthreads();
    }
}

// ---------------------------------------------------------------------------
// Launch
// ---------------------------------------------------------------------------
extern "C" void kernel_launch(void* const* d_in, const int* in_sizes, int n_in,
                              void* d_out, int out_size, void* d_ws, size_t ws_size,
                              hipStream_t stream) {
    const float* x       = (const float*)d_in[0];   // (256,)
    const float* actives = (const float*)d_in[1];   // (2048, 832)
    const float* weights = (const float*)d_in[2];   // (512, 32, 3)
    const int*   in_idxs = (const int*)d_in[3];     // (512, 32)
    float* outp = (float*)d_out;                    // (64,)

    float* ws  = (float*)d_ws;
    float* AsT = ws;                                 // 832*2048  = 1,703,936 f
    float* B   = AsT + (size_t)D_LEN * T_LEN;        // 832*1024  =   851,968 f
    float* KV  = B + (size_t)D_LEN * NCOLS;          // 2048*1024 = 2,097,152 f
    // total ~18.6 MB of workspace

    prep_ast_kernel<<<(T_LEN * D_LEN) / 256, 256, 0, stream>>>(actives, AsT);
    zero_b_kernel<<<(D_LEN * NCOLS) / 256, 256, 0, stream>>>(B);
    scatter_w_kernel<<<(N_NODES + 255) / 256, 256, 0, stream>>>(weights, in_idxs, B);

    // GEMM: (2048/64) M-blocks x (1024/64) N-blocks = 512 blocks, 4 waves each
    gemm_kv_kernel<<<(T_LEN / 64) * (NCOLS / 64), 128, 0, stream>>>(AsT, B, KV);

    seq_attn_kernel<<<1, 1024, 0, stream>>>(x, weights, in_idxs, KV, outp);
}